// GMDSSL_72121090834751
// MI455X (gfx1250) — compile-verified
//
#include <hip/hip_runtime.h>
#include <math.h>

// Problem constants (from reference)
#define Bb 16
#define Tt 12
#define Nn 325
#define Cc 64
#define Kk 8
#define Ll (Nn * Tt)   // 3900
#define CLr (Cc * Ll)  // 249600  (gamma reduction length)
#define HALF_LOG2PI 0.9189385332046727f

typedef __attribute__((ext_vector_type(2))) float v2f;
typedef __attribute__((ext_vector_type(8))) float v8f;

// ---------------------------------------------------------------------------
// Kernel 1: l2-normalize rep / rep_aug over C and transpose [b,t,n,c]->[b][c][l]
// (l = n*T + t). One wave32 per (b,t,n) row; lane handles c and c+32.
// ---------------------------------------------------------------------------
__global__ __launch_bounds__(256) void norm_kernel(const float* __restrict__ rep,
                                                   const float* __restrict__ rep_aug,
                                                   float* __restrict__ Hn,
                                                   float* __restrict__ An) {
    int tid  = threadIdx.x;
    int warp = tid >> 5;
    int lane = tid & 31;
    int row  = blockIdx.x * 8 + warp;  // (b,t,n) flat index
    if (row >= Bb * Tt * Nn) return;
    int b   = row / (Tt * Nn);
    int rem = row - b * (Tt * Nn);
    int t   = rem / Nn;
    int n   = rem - t * Nn;
    int l   = n * Tt + t;

    const float* pr = rep     + (size_t)row * Cc;
    const float* pa = rep_aug + (size_t)row * Cc;
    float r0 = pr[lane], r1 = pr[lane + 32];
    float a0 = pa[lane], a1 = pa[lane + 32];
    float sr = r0 * r0 + r1 * r1;
    float sa = a0 * a0 + a1 * a1;
    for (int off = 16; off > 0; off >>= 1) {
        sr += __shfl_xor(sr, off, 32);
        sa += __shfl_xor(sa, off, 32);
    }
    float rr = 1.0f / fmaxf(sqrtf(sr), 1e-12f);
    float ra = 1.0f / fmaxf(sqrtf(sa), 1e-12f);
    size_t base = (size_t)b * CLr + l;
    Hn[base + (size_t)lane * Ll]        = r0 * rr;
    Hn[base + (size_t)(lane + 32) * Ll] = r1 * rr;
    An[base + (size_t)lane * Ll]        = a0 * ra;
    An[base + (size_t)(lane + 32) * Ll] = a1 * ra;
}

// ---------------------------------------------------------------------------
// Kernel 2: gamma logits via V_WMMA_F32_16X16X4_F32.
// logits[b,k] = sum_r An[b][r] * W_gamma[k][r], r = c*L + l (matches [b][c][l]).
// A: M=16 batches, B: N=16 (k padded 8->16), split-K across 96 waves.
// Padding lanes (k>=8) read a zeroed scratch word with stride 0: every lane
// issues an unconditional load -> no exec-mask branches in the hot loop and
// EXEC stays all-ones around every WMMA.
// ---------------------------------------------------------------------------
#define GAMMA_WGS 96
#define GAMMA_STEPS (CLr / 4 / GAMMA_WGS)  // 650, exact

__global__ __launch_bounds__(32) void gamma_wmma_kernel(const float* __restrict__ An,
                                                        const float* __restrict__ Wg,
                                                        const float* __restrict__ zpad,
                                                        float* __restrict__ logits) {
    int lane = threadIdx.x;
    int m    = lane & 15;
    int hi   = lane >> 4;
    int rbase = blockIdx.x * GAMMA_STEPS * 4;

    const float* pa = An + (size_t)m * CLr + rbase + 2 * hi;  // A row = batch m
    bool bvalid = (m < Kk);
    const float* pb = bvalid ? (Wg + (size_t)m * CLr + rbase + 2 * hi) : zpad;
    const int    sb = bvalid ? 4 : 0;  // padding lanes re-read the zero word

    v8f acc = {};
    for (int i = 0; i < GAMMA_STEPS; ++i) {
        v2f a  = *(const v2f*)pa;
        v2f wb = *(const v2f*)pb;
        __builtin_prefetch(pa + 64, 0, 3);
        acc = __builtin_amdgcn_wmma_f32_16x16x4_f32(false, a, false, wb,
                                                    (short)0, acc, false, false);
        pa += 4; pb += sb;
    }
    // D layout: VGPR v holds rows M = v + 8*hi, column N = lane&15
    for (int v = 0; v < 8; ++v) {
        int rowm = v + 8 * hi;
        atomicAdd(&logits[rowm * 16 + m], acc[v]);
    }
}

// ---------------------------------------------------------------------------
// Kernel 3: log-softmax of logits over K -> loggam[b][k]
// ---------------------------------------------------------------------------
__global__ void gamma_finalize_kernel(const float* __restrict__ logits,
                                      float* __restrict__ loggam) {
    int b = threadIdx.x;
    if (b >= Bb) return;
    float x[Kk];
    float mx = -1e30f;
    for (int k = 0; k < Kk; ++k) { x[k] = logits[b * 16 + k]; mx = fmaxf(mx, x[k]); }
    float s = 0.0f;
    for (int k = 0; k < Kk; ++k) s += expf(x[k] - mx);
    float lse = mx + logf(s);
    for (int k = 0; k < Kk; ++k) loggam[b * Kk + k] = x[k] - lse;
}

// ---------------------------------------------------------------------------
// Kernel 4: mu/sigma GEMMs per batch via WMMA.
// mus_raw[b,c,k] = sum_l An[b][c][l] * Wmu[k][l]  (A rows = channels, N = k).
// Both mu and sigma accumulators reuse the same A-tile load.
// Finalize: mu = raw+bmu ; inv_sigma = exp(-(raw+bsig)) ; nls = -(raw+bsig)-0.5*log(2pi)
// ---------------------------------------------------------------------------
#define MS_STEPS (Ll / 4)  // 975, exact

__global__ __launch_bounds__(32) void musig_wmma_kernel(const float* __restrict__ An,
                                                        const float* __restrict__ Wmu,
                                                        const float* __restrict__ Wsig,
                                                        const float* __restrict__ bmu,
                                                        const float* __restrict__ bsig,
                                                        const float* __restrict__ zpad,
                                                        float* __restrict__ mu,
                                                        float* __restrict__ invs,
                                                        float* __restrict__ nls) {
    int lane = threadIdx.x;
    int m    = lane & 15;
    int hi   = lane >> 4;
    int b    = blockIdx.x >> 2;
    int ct   = blockIdx.x & 3;
    int c    = ct * 16 + m;

    const float* pa = An + ((size_t)b * Cc + c) * Ll + 2 * hi;
    bool bvalid = (m < Kk);
    const float* pm = bvalid ? (Wmu  + (size_t)m * Ll + 2 * hi) : zpad;
    const float* ps = bvalid ? (Wsig + (size_t)m * Ll + 2 * hi) : zpad;
    const int    sb = bvalid ? 4 : 0;

    v8f amu = {}, asg = {};
    for (int i = 0; i < MS_STEPS; ++i) {
        v2f a  = *(const v2f*)pa;
        v2f wm = *(const v2f*)pm;
        v2f ws = *(const v2f*)ps;
        __builtin_prefetch(pa + 64, 0, 3);
        amu = __builtin_amdgcn_wmma_f32_16x16x4_f32(false, a, false, wm,
                                                    (short)0, amu, false, false);
        asg = __builtin_amdgcn_wmma_f32_16x16x4_f32(false, a, false, ws,
                                                    (short)0, asg, false, false);
        pa += 4; pm += sb; ps += sb;
    }
    if (bvalid) {  // this lane's D column n = m is a real k index
        float bmv = bmu[m], bsv = bsig[m];
        for (int v = 0; v < 8; ++v) {
            int cc = ct * 16 + v + 8 * hi;  // D row = channel within batch
            float mr = amu[v] + bmv;
            float sr = asg[v] + bsv;        // = log(sigma)
            size_t idx = ((size_t)b * Cc + cc) * Kk + m;
            mu[idx]   = mr;
            invs[idx] = expf(-sr);
            nls[idx]  = -sr - HALF_LOG2PI;
        }
    }
}

// ---------------------------------------------------------------------------
// Kernel 5: P[b,l,k] = prod_c ( nls[b,c,k] - 0.5*((h-mu)*inv_sigma)^2 )
// Block handles (b, 32-l tile); h tile + per-b params staged in LDS.
// Also accumulates sum over l of P^2 per (b,k) (for the l2-norm over L).
// ---------------------------------------------------------------------------
#define LT 32
#define LTILES ((Ll + LT - 1) / LT)  // 122

__global__ __launch_bounds__(256) void logp_kernel(const float* __restrict__ Hn,
                                                   const float* __restrict__ mu,
                                                   const float* __restrict__ invs,
                                                   const float* __restrict__ nls,
                                                   float* __restrict__ P,
                                                   float* __restrict__ sumsq) {
    __shared__ float hs[Cc * LT];
    __shared__ float ms[Cc * Kk], iss[Cc * Kk], ns[Cc * Kk];
    __shared__ float ksum[Kk];
    int tid  = threadIdx.x;
    int b    = blockIdx.x / LTILES;
    int tile = blockIdx.x - b * LTILES;
    int l0   = tile * LT;

    for (int idx = tid; idx < Cc * LT; idx += 256) {
        int c = idx >> 5, ll = idx & 31;
        int l = l0 + ll;
        hs[idx] = (l < Ll) ? Hn[((size_t)b * Cc + c) * Ll + l] : 0.0f;
    }
    for (int idx = tid; idx < Cc * Kk; idx += 256) {
        size_t g = (size_t)b * Cc * Kk + idx;
        ms[idx] = mu[g]; iss[idx] = invs[g]; ns[idx] = nls[g];
    }
    if (tid < Kk) ksum[tid] = 0.0f;
    __syncthreads();

    int ll = tid >> 3;
    int k  = tid & 7;
    int l  = l0 + ll;
    if (l < Ll) {
        float p = 1.0f;
#pragma unroll 8
        for (int c = 0; c < Cc; ++c) {
            float z = (hs[c * LT + ll] - ms[c * Kk + k]) * iss[c * Kk + k];
            p *= (ns[c * Kk + k] - 0.5f * z * z);
        }
        P[((size_t)b * Ll + l) * Kk + k] = p;
        atomicAdd(&ksum[k], p * p);
    }
    __syncthreads();
    if (tid < Kk) atomicAdd(&sumsq[b * Kk + tid], ksum[tid]);
}

// ---------------------------------------------------------------------------
// Kernel 6: loss = -mean_{b,l} log( sum_k exp( P/||P||_L + loggam ) )
// ---------------------------------------------------------------------------
__global__ __launch_bounds__(256) void final_kernel(const float* __restrict__ P,
                                                    const float* __restrict__ sumsq,
                                                    const float* __restrict__ loggam,
                                                    float* __restrict__ out) {
    __shared__ float red[256];
    int i = blockIdx.x * 256 + threadIdx.x;
    float part = 0.0f;
    if (i < Bb * Ll) {
        int b = i / Ll;
        float s = 0.0f;
        for (int k = 0; k < Kk; ++k) {
            float rn = 1.0f / fmaxf(sqrtf(sumsq[b * Kk + k]), 1e-12f);
            float lp = P[(size_t)i * Kk + k] * rn + loggam[b * Kk + k];
            s += expf(lp);
        }
        part = -logf(s);
    }
    red[threadIdx.x] = part;
    __syncthreads();
    for (int off = 128; off > 0; off >>= 1) {
        if (threadIdx.x < off) red[threadIdx.x] += red[threadIdx.x + off];
        __syncthreads();
    }
    if (threadIdx.x == 0) atomicAdd(out, red[0] * (1.0f / (float)(Bb * Ll)));
}

// ---------------------------------------------------------------------------
extern "C" void kernel_launch(void* const* d_in, const int* in_sizes, int n_in,
                              void* d_out, int out_size, void* d_ws, size_t ws_size,
                              hipStream_t stream) {
    (void)in_sizes; (void)n_in; (void)out_size; (void)ws_size;
    const float* rep  = (const float*)d_in[0];
    const float* repa = (const float*)d_in[1];
    const float* Wg   = (const float*)d_in[2];
    const float* Wmu  = (const float*)d_in[3];
    const float* bmu  = (const float*)d_in[4];
    const float* Wsig = (const float*)d_in[5];
    const float* bsig = (const float*)d_in[6];
    float* out = (float*)d_out;

    // Workspace carve-out (~34 MB total of fp32)
    float* w = (float*)d_ws;
    size_t off = 0;
    float* Hn     = w + off; off += (size_t)Bb * CLr;       // 3,993,600
    float* An     = w + off; off += (size_t)Bb * CLr;       // 3,993,600
    float* P      = w + off; off += (size_t)Bb * Ll * Kk;   //   499,200
    float* mu     = w + off; off += (size_t)Bb * Cc * Kk;   //     8,192
    float* invs   = w + off; off += (size_t)Bb * Cc * Kk;
    float* nls    = w + off; off += (size_t)Bb * Cc * Kk;
    float* logits = w + off; off += 16 * 16;
    float* loggam = w + off; off += Bb * Kk;
    float* sumsq  = w + off; off += Bb * Kk;
    float* zpad   = w + off; off += 8;   // zero pad for k>=8 WMMA B-columns

    // Zero the accumulator staging each call (graph-capture safe)
    hipMemsetAsync(logits, 0, 16 * 16 * sizeof(float), stream);
    hipMemsetAsync(sumsq, 0, Bb * Kk * sizeof(float), stream);
    hipMemsetAsync(zpad, 0, 8 * sizeof(float), stream);
    hipMemsetAsync(out, 0, sizeof(float), stream);

    norm_kernel<<<(Bb * Tt * Nn + 7) / 8, 256, 0, stream>>>(rep, repa, Hn, An);
    gamma_wmma_kernel<<<GAMMA_WGS, 32, 0, stream>>>(An, Wg, zpad, logits);
    gamma_finalize_kernel<<<1, 32, 0, stream>>>(logits, loggam);
    musig_wmma_kernel<<<Bb * 4, 32, 0, stream>>>(An, Wmu, Wsig, bmu, bsig, zpad,
                                                 mu, invs, nls);
    logp_kernel<<<Bb * LTILES, 256, 0, stream>>>(Hn, mu, invs, nls, P, sumsq);
    final_kernel<<<(Bb * Ll + 255) / 256, 256, 0, stream>>>(P, sumsq, loggam, out);
}